// GEMTEGraph3DMPNN_21414706938038
// MI455X (gfx1250) — compile-verified
//
#include <hip/hip_runtime.h>

// FDTD 3D, two-pass leapfrog update. Memory-bound: ~425 MB compulsory traffic
// at 23.3 TB/s => ~18-25 us. Strategy: coalesced b32 streams along z,
// H^{n+1/2} intermediate (85 MB) kept L2-resident (192 MB L2) between passes,
// NT hints on single-use data, global_prefetch_b8 stream-ahead.

#define NX 192
#define NY 192
#define NZ 192

__device__ __constant__ float kInv2dx = 500.0f;   // 1/(2*1e-3)
__device__ __constant__ float kInv2dy = 500.0f;
__device__ __constant__ float kInv2dz = 500.0f;
#define DT_F 1.0e-12f

// ---------------- Pass 1: H^{n+1/2} = H - dt/mu * curl(E) ----------------
__global__ __launch_bounds__(NZ) void fdtd_step1(
    const float* __restrict__ Ex, const float* __restrict__ Ey,
    const float* __restrict__ Ez, const float* __restrict__ Hx,
    const float* __restrict__ Hy, const float* __restrict__ Hz,
    const float* __restrict__ mu,
    float* __restrict__ Hx2, float* __restrict__ Hy2, float* __restrict__ Hz2)
{
    const int k = threadIdx.x;      // contiguous
    const int j = blockIdx.y;
    const int i = blockIdx.x;
    const int sj = NZ;              // j stride
    const int si = NY * NZ;         // i stride
    const int idx = (i * NY + j) * NZ + k;

    // Prefetch the x-plane two ahead (i+2) of the E fields we stream (gfx1250 global_prefetch_b8).
    if (i + 2 < NX) {
        __builtin_prefetch(Ex + idx + 2 * si, 0, 3);
        __builtin_prefetch(Ey + idx + 2 * si, 0, 3);
        __builtin_prefetch(Ez + idx + 2 * si, 0, 3);
    }

    float curlx = 0.0f, curly = 0.0f, curlz = 0.0f;
    const bool interior = (i >= 1) & (i < NX - 1) & (j >= 1) & (j < NY - 1) &
                          (k >= 1) & (k < NZ - 1);
    if (interior) {
        const float dEz_dy = (Ez[idx + sj] - Ez[idx - sj]) * kInv2dy;
        const float dEy_dz = (Ey[idx + 1]  - Ey[idx - 1])  * kInv2dz;
        const float dEx_dz = (Ex[idx + 1]  - Ex[idx - 1])  * kInv2dz;
        const float dEz_dx = (Ez[idx + si] - Ez[idx - si]) * kInv2dx;
        const float dEy_dx = (Ey[idx + si] - Ey[idx - si]) * kInv2dx;
        const float dEx_dy = (Ex[idx + sj] - Ex[idx - sj]) * kInv2dy;
        curlx = dEz_dy - dEy_dz;
        curly = dEx_dz - dEz_dx;
        curlz = dEy_dx - dEx_dy;
    }

    // mu and H are read exactly once in the whole problem -> non-temporal loads
    // (don't evict the E fields / H2 outputs that pass 2 will re-read from L2).
    const float s  = DT_F / __builtin_nontemporal_load(mu + idx);
    const float hx = __builtin_nontemporal_load(Hx + idx);
    const float hy = __builtin_nontemporal_load(Hy + idx);
    const float hz = __builtin_nontemporal_load(Hz + idx);

    // H2 is re-read by pass 2 -> regular (L2-resident) stores.
    Hx2[idx] = fmaf(-s, curlx, hx);
    Hy2[idx] = fmaf(-s, curly, hy);
    Hz2[idx] = fmaf(-s, curlz, hz);
}

// ---- Pass 2: E^{n+1} = (A-/A+) E + dt/(eps*A+) * curl(H^{n+1/2}) ----
__global__ __launch_bounds__(NZ) void fdtd_step2(
    const float* __restrict__ Ex, const float* __restrict__ Ey,
    const float* __restrict__ Ez, const float* __restrict__ Hx2,
    const float* __restrict__ Hy2, const float* __restrict__ Hz2,
    const float* __restrict__ eps, const float* __restrict__ sigma,
    float* __restrict__ Ex2, float* __restrict__ Ey2, float* __restrict__ Ez2)
{
    const int k = threadIdx.x;
    const int j = blockIdx.y;
    const int i = blockIdx.x;
    const int sj = NZ;
    const int si = NY * NZ;
    const int idx = (i * NY + j) * NZ + k;

    if (i + 2 < NX) {
        __builtin_prefetch(Hx2 + idx + 2 * si, 0, 3);
        __builtin_prefetch(Hy2 + idx + 2 * si, 0, 3);
        __builtin_prefetch(Hz2 + idx + 2 * si, 0, 3);
    }

    float curlx = 0.0f, curly = 0.0f, curlz = 0.0f;
    const bool interior = (i >= 1) & (i < NX - 1) & (j >= 1) & (j < NY - 1) &
                          (k >= 1) & (k < NZ - 1);
    if (interior) {
        const float dHz_dy = (Hz2[idx + sj] - Hz2[idx - sj]) * kInv2dy;
        const float dHy_dz = (Hy2[idx + 1]  - Hy2[idx - 1])  * kInv2dz;
        const float dHx_dz = (Hx2[idx + 1]  - Hx2[idx - 1])  * kInv2dz;
        const float dHz_dx = (Hz2[idx + si] - Hz2[idx - si]) * kInv2dx;
        const float dHy_dx = (Hy2[idx + si] - Hy2[idx - si]) * kInv2dx;
        const float dHx_dy = (Hx2[idx + sj] - Hx2[idx - sj]) * kInv2dy;
        curlx = dHz_dy - dHy_dz;
        curly = dHx_dz - dHz_dx;
        curlz = dHy_dx - dHx_dy;
    }

    // eps/sigma/E read once here -> non-temporal loads.
    const float ep = __builtin_nontemporal_load(eps + idx);
    const float sg = __builtin_nontemporal_load(sigma + idx);
    const float ex = __builtin_nontemporal_load(Ex + idx);
    const float ey = __builtin_nontemporal_load(Ey + idx);
    const float ez = __builtin_nontemporal_load(Ez + idx);

    const float half_sdt = sg * (DT_F * 0.5f) / ep;   // sigma*dt/(2*eps)
    const float a_plus   = 1.0f + half_sdt;
    const float inv_ap   = 1.0f / a_plus;
    const float coefE    = (1.0f - half_sdt) * inv_ap;
    const float coefC    = (DT_F / ep) * inv_ap;

    // Final outputs, never re-read on device -> non-temporal stores.
    __builtin_nontemporal_store(fmaf(coefC, curlx, coefE * ex), Ex2 + idx);
    __builtin_nontemporal_store(fmaf(coefC, curly, coefE * ey), Ey2 + idx);
    __builtin_nontemporal_store(fmaf(coefC, curlz, coefE * ez), Ez2 + idx);
}

extern "C" void kernel_launch(void* const* d_in, const int* in_sizes, int n_in,
                              void* d_out, int out_size, void* d_ws, size_t ws_size,
                              hipStream_t stream) {
    // setup_inputs order: Ex, Ey, Ez, Hx, Hy, Hz, eps, mu, sigma
    const float* Ex    = (const float*)d_in[0];
    const float* Ey    = (const float*)d_in[1];
    const float* Ez    = (const float*)d_in[2];
    const float* Hx    = (const float*)d_in[3];
    const float* Hy    = (const float*)d_in[4];
    const float* Hz    = (const float*)d_in[5];
    const float* eps   = (const float*)d_in[6];
    const float* mu    = (const float*)d_in[7];
    const float* sigma = (const float*)d_in[8];

    const int n = NX * NY * NZ;
    float* out = (float*)d_out;
    float* Ex2 = out + 0 * n;
    float* Ey2 = out + 1 * n;
    float* Ez2 = out + 2 * n;
    float* Hx2 = out + 3 * n;
    float* Hy2 = out + 4 * n;
    float* Hz2 = out + 5 * n;

    dim3 grid(NX, NY, 1);
    dim3 block(NZ, 1, 1);   // 192 threads = 6 wave32, k contiguous

    fdtd_step1<<<grid, block, 0, stream>>>(Ex, Ey, Ez, Hx, Hy, Hz, mu,
                                           Hx2, Hy2, Hz2);
    fdtd_step2<<<grid, block, 0, stream>>>(Ex, Ey, Ez, Hx2, Hy2, Hz2,
                                           eps, sigma, Ex2, Ey2, Ez2);
}